// GINEGCN_15324443312656
// MI455X (gfx1250) — compile-verified
//
#include <hip/hip_runtime.h>
#include <math.h>

#define NN   256   // nodes
#define HID  128
#define LAY  4
#define NCAT 64
#define BB   8
#define ROWS (BB*NN)   // 2048

typedef __attribute__((ext_vector_type(2))) float v2f;
typedef __attribute__((ext_vector_type(8))) float v8f;

__device__ __forceinline__ float red16(float v) {
    // reduce across 16-lane half-wave (masks 1,2,4,8 stay inside each half)
    #pragma unroll
    for (int m = 1; m < 16; m <<= 1) v += __shfl_xor(v, m, 32);
    return v;
}
__device__ __forceinline__ float red32(float v) {
    #pragma unroll
    for (int m = 1; m < 32; m <<= 1) v += __shfl_xor(v, m, 32);
    return v;
}

// ---------------- kernel 1: per-category max-norm clip of emb -> embn ----------------
__global__ __launch_bounds__(32) void gine_embnorm(const float* __restrict__ emb,
                                                   float* __restrict__ embn) {
    int c = blockIdx.x;          // 0..63
    int lane = threadIdx.x;      // 0..31
    float v[4]; float ss = 0.f;
    #pragma unroll
    for (int q = 0; q < 4; ++q) {
        v[q] = emb[c * HID + lane * 4 + q];
        ss += v[q] * v[q];
    }
    ss = red32(ss);
    float n = sqrtf(ss);
    float scale = (n > 1.f) ? (1.f / n) : 1.f;   // e * min(1, 1/||e||), 0-norm -> 1
    #pragma unroll
    for (int q = 0; q < 4; ++q) embn[c * HID + lane * 4 + q] = v[q] * scale;
}

// ---------------- kernel 2: t = relu(LN(x @ in_W1 + b1)) (K = CIN = 2) ----------------
__global__ __launch_bounds__(256) void gine_in1(const float* __restrict__ x,
                                                const float* __restrict__ W1,
                                                const float* __restrict__ b1,
                                                const float* __restrict__ g,
                                                const float* __restrict__ bt,
                                                float* __restrict__ out) {
    int row = blockIdx.x * 16 + (threadIdx.x >> 4);  // global row b*N+n
    int l16 = threadIdx.x & 15;
    float x0 = x[row * 2 + 0], x1 = x[row * 2 + 1];
    float v[8]; float s = 0.f;
    #pragma unroll
    for (int q = 0; q < 8; ++q) {
        int c = l16 * 8 + q;
        float h = x0 * W1[c] + x1 * W1[HID + c] + b1[c];
        v[q] = h; s += h;
    }
    s = red16(s);
    float mean = s * (1.f / HID);
    float ss = 0.f;
    #pragma unroll
    for (int q = 0; q < 8; ++q) { float d = v[q] - mean; ss += d * d; }
    ss = red16(ss);
    float inv = rsqrtf(ss * (1.f / HID) + 1e-5f);
    #pragma unroll
    for (int q = 0; q < 8; ++q) {
        int c = l16 * 8 + q;
        float y = (v[q] - mean) * inv * g[c] + bt[c];
        out[row * HID + c] = fmaxf(y, 0.f);
    }
}

// ------------- kernel 3: WMMA GEMM [16x128]x[128x128] + fused LayerNorm epilogues ----
// mode 0: out = LN(A@W + bias; g1,bt1)
// mode 1: out = relu(LN(A@W + bias; g1,bt1))
// mode 2: t = LN(A@W + bias; g1,bt1); out = relu(LN(t; g2,bt2)) + resid
__global__ __launch_bounds__(256) void gine_mm_ln(const float* __restrict__ A,
                                                  const float* __restrict__ W,
                                                  const float* __restrict__ bias,
                                                  const float* __restrict__ g1,
                                                  const float* __restrict__ bt1,
                                                  const float* __restrict__ g2,
                                                  const float* __restrict__ bt2,
                                                  const float* __restrict__ resid,
                                                  float* __restrict__ out,
                                                  int mode) {
    __shared__ float As[16][132];   // +4 pad: bank = (4*row + k) % 64, conflict-free
    __shared__ float Cs[16][132];
    int t = threadIdx.x;
    int rowBase = blockIdx.x * 16;

    for (int k = t; k < 16 * HID; k += 256) {
        int r = k >> 7, c = k & 127;
        As[r][c] = A[(rowBase + r) * HID + c];
    }
    __syncthreads();

    int wave = t >> 5;
    int lane = t & 31;
    int l16  = lane & 15;
    int hi   = lane >> 4;
    int colBase = wave * 16;     // 8 waves cover 128 cols

    v8f acc = {0.f, 0.f, 0.f, 0.f, 0.f, 0.f, 0.f, 0.f};
    #pragma unroll
    for (int k0 = 0; k0 < HID; k0 += 4) {
        int ka = k0 + 2 * hi;                      // A lane t: row=l16, k = 2*hi + v
        v2f a; a.x = As[l16][ka]; a.y = As[l16][ka + 1];
        v2f b;                                     // B lane t: col=l16, k = 2*hi + v
        b.x = W[ka * HID + colBase + l16];
        b.y = W[(ka + 1) * HID + colBase + l16];
        acc = __builtin_amdgcn_wmma_f32_16x16x4_f32(
            false, a, false, b, (short)0, acc, false, false);
    }

    // C/D layout: reg r -> row r + 8*hi, col l16
    #pragma unroll
    for (int r = 0; r < 8; ++r) {
        int rr = r + 8 * hi;
        int cc = colBase + l16;
        Cs[rr][cc] = acc[r] + bias[cc];
    }
    __syncthreads();

    // cooperative LayerNorm: 16 threads per row, 8 cols each, two-pass
    int row  = t >> 4;
    int cgrp = t & 15;
    float v[8]; float s = 0.f;
    #pragma unroll
    for (int q = 0; q < 8; ++q) { v[q] = Cs[row][cgrp * 8 + q]; s += v[q]; }
    s = red16(s);
    float mean = s * (1.f / HID);
    float ss = 0.f;
    #pragma unroll
    for (int q = 0; q < 8; ++q) { float d = v[q] - mean; ss += d * d; }
    ss = red16(ss);
    float inv = rsqrtf(ss * (1.f / HID) + 1e-5f);

    int gRow = rowBase + row;
    if (mode == 2) {
        float tv[8]; float s2 = 0.f;
        #pragma unroll
        for (int q = 0; q < 8; ++q) {
            int c = cgrp * 8 + q;
            tv[q] = (v[q] - mean) * inv * g1[c] + bt1[c];
            s2 += tv[q];
        }
        s2 = red16(s2);
        float m2 = s2 * (1.f / HID);
        float ss2 = 0.f;
        #pragma unroll
        for (int q = 0; q < 8; ++q) { float d = tv[q] - m2; ss2 += d * d; }
        ss2 = red16(ss2);
        float inv2 = rsqrtf(ss2 * (1.f / HID) + 1e-5f);
        #pragma unroll
        for (int q = 0; q < 8; ++q) {
            int c = cgrp * 8 + q;
            float y = fmaxf((tv[q] - m2) * inv2 * g2[c] + bt2[c], 0.f)
                      + resid[gRow * HID + c];
            out[gRow * HID + c] = y;
        }
    } else {
        #pragma unroll
        for (int q = 0; q < 8; ++q) {
            int c = cgrp * 8 + q;
            float y = (v[q] - mean) * inv * g1[c] + bt1[c];
            if (mode == 1) y = fmaxf(y, 0.f);
            out[gRow * HID + c] = y;
        }
    }
}

// -------- kernel 4: u[b,j,:] = (1+eps_l)*h[b,j,:] + sum_i relu(h[b,i,:]+embn[cat(i,j)]) --
__global__ __launch_bounds__(128) void gine_agg(const float* __restrict__ h,
                                                const float* __restrict__ embn,
                                                const int* __restrict__ cat,
                                                const float* __restrict__ eps_arr,
                                                int l,
                                                float* __restrict__ u) {
    int bj = blockIdx.x;            // b*N + j
    int b = bj >> 8, j = bj & 255;
    int d = threadIdx.x;            // 0..127
    const float* hb = h + (size_t)b * NN * HID;
    float acc = 0.f;
    for (int i = 0; i < NN; ++i) {
        int c = cat[i * NN + j];                 // edge (i,j) at index i*N+j
        float v = hb[i * HID + d] + embn[c * HID + d];
        acc += fmaxf(v, 0.f);
    }
    float e = eps_arr[l];
    u[(size_t)bj * HID + d] = (1.f + e) * hb[j * HID + d] + acc;
}

// ---------------- kernel 5: out = h @ out_W + out_b  (128 -> 3) ----------------
__global__ __launch_bounds__(256) void gine_out(const float* __restrict__ h,
                                                const float* __restrict__ W,
                                                const float* __restrict__ bias,
                                                float* __restrict__ out) {
    int row = blockIdx.x * 8 + (threadIdx.x >> 5);   // one wave per row
    int lane = threadIdx.x & 31;
    float a0 = 0.f, a1 = 0.f, a2 = 0.f;
    #pragma unroll
    for (int q = 0; q < 4; ++q) {
        int d = lane * 4 + q;
        float hv = h[row * HID + d];
        a0 += hv * W[d * 3 + 0];
        a1 += hv * W[d * 3 + 1];
        a2 += hv * W[d * 3 + 2];
    }
    a0 = red32(a0); a1 = red32(a1); a2 = red32(a2);
    if (lane == 0) {
        out[row * 3 + 0] = a0 + bias[0];
        out[row * 3 + 1] = a1 + bias[1];
        out[row * 3 + 2] = a2 + bias[2];
    }
}

extern "C" void kernel_launch(void* const* d_in, const int* in_sizes, int n_in,
                              void* d_out, int out_size, void* d_ws, size_t ws_size,
                              hipStream_t stream) {
    const float* x       = (const float*)d_in[0];
    const float* emb     = (const float*)d_in[1];
    const float* in_W1   = (const float*)d_in[2];
    const float* in_b1   = (const float*)d_in[3];
    const float* in_g1   = (const float*)d_in[4];
    const float* in_bt1  = (const float*)d_in[5];
    const float* in_W2   = (const float*)d_in[6];
    const float* in_b2   = (const float*)d_in[7];
    const float* in_g2   = (const float*)d_in[8];
    const float* in_bt2  = (const float*)d_in[9];
    const float* conv_W1 = (const float*)d_in[10];
    const float* conv_b1 = (const float*)d_in[11];
    const float* conv_g1 = (const float*)d_in[12];
    const float* conv_bt1= (const float*)d_in[13];
    const float* conv_W2 = (const float*)d_in[14];
    const float* conv_b2 = (const float*)d_in[15];
    const float* conv_g2 = (const float*)d_in[16];
    const float* conv_bt2= (const float*)d_in[17];
    const float* conv_eps= (const float*)d_in[18];
    const float* ln_g    = (const float*)d_in[19];
    const float* ln_b    = (const float*)d_in[20];
    const float* out_W   = (const float*)d_in[21];
    const float* out_b   = (const float*)d_in[22];
    const int*   edge_cat= (const int*)d_in[25];
    float* out = (float*)d_out;

    float* ws   = (float*)d_ws;
    float* embn = ws;                         // 64*128
    float* bufA = embn + NCAT * HID;          // 2048*128 (MLP hidden scratch)
    float* bufH = bufA + ROWS * HID;          // 2048*128 (node state h)
    float* bufU = bufH + ROWS * HID;          // 2048*128 (aggregated input u)

    gine_embnorm<<<NCAT, 32, 0, stream>>>(emb, embn);
    gine_in1<<<ROWS / 16, 256, 0, stream>>>(x, in_W1, in_b1, in_g1, in_bt1, bufA);
    gine_mm_ln<<<ROWS / 16, 256, 0, stream>>>(bufA, in_W2, in_b2, in_g2, in_bt2,
                                              nullptr, nullptr, nullptr, bufH, 0);
    for (int l = 0; l < LAY; ++l) {
        gine_agg<<<ROWS, 128, 0, stream>>>(bufH, embn, edge_cat, conv_eps, l, bufU);
        gine_mm_ln<<<ROWS / 16, 256, 0, stream>>>(
            bufU, conv_W1 + l * HID * HID, conv_b1 + l * HID,
            conv_g1 + l * HID, conv_bt1 + l * HID,
            nullptr, nullptr, nullptr, bufA, 1);
        gine_mm_ln<<<ROWS / 16, 256, 0, stream>>>(
            bufA, conv_W2 + l * HID * HID, conv_b2 + l * HID,
            conv_g2 + l * HID, conv_bt2 + l * HID,
            ln_g + l * HID, ln_b + l * HID, bufH, bufH, 2);
    }
    gine_out<<<ROWS / 8, 256, 0, stream>>>(bufH, out_W, out_b, out);
}